// DepthPredictorCrossMultiView_22823456211415
// MI455X (gfx1250) — compile-verified
//
#include <hip/hip_runtime.h>
#include <math.h>

// ---------------------------------------------------------------------------
// Problem constants (fixed by setup_inputs)
// ---------------------------------------------------------------------------
#define B_  2
#define C_  128
#define H_  64
#define W_  128
#define D_  32
#define HW_ (H_ * W_)

#define PHI_MIN_F   (-2.356194490192345f)   // -3*pi/4
#define PHI_RNG_F   ( 4.712388980384690f)   //  3*pi/2
#define THETA_MIN_F ( 0.785398163397448f)   //  pi/4
#define THETA_RNG_F ( 1.570796326794897f)   //  pi/2

typedef float v2f __attribute__((ext_vector_type(2)));
typedef float v8f __attribute__((ext_vector_type(8)));
typedef float f2  __attribute__((ext_vector_type(2)));

// ---------------------------------------------------------------------------
// Kernel 1: rotate direction vectors with V_WMMA_F32_16X16X4_F32.
//   rot_yin  = R        @ dirs      (3 x HW per batch)
//   rot_yang = (R @ M)  @ dirs      (M = yang hemisphere swap)
// A-matrix (16x4 f32): lanes 0-15 hold {K=0,K=1} in v[0],v[1];
//                      lanes 16-31 hold {K=2,K=3}.  Rows m>=3 are zero.
// B-matrix (4x16 f32): mirrored layout, lane n holds column n.
// D (16x16 f32): lane n (<16), vgpr r = D[m=r][n]; rows 0..2 are the result.
// One wave handles 16 directions; B*HW/16 = 1024 waves total.
// Workspace layout: rot[b][view(2)][comp(3)][HW]  (float)
// ---------------------------------------------------------------------------
__global__ __launch_bounds__(32)
void rot_dirs_wmma_kernel(const float* __restrict__ pose,
                          float* __restrict__ rot)
{
    const int lane = threadIdx.x;          // 0..31, full wave
    const int blk  = blockIdx.x;           // 0 .. B*HW/16 - 1
    const int b    = blk / (HW_ / 16);
    const int tile = blk % (HW_ / 16);
    const int m    = lane & 15;
    const bool hi  = lane >= 16;
    const int n    = tile * 16 + m;        // direction index (h*W + w)

    // direction vector for pixel n (every lane computes its own copy)
    const float gx = (float)(n % W_);
    const float gy = (float)(n / W_);
    const float phi   = gx * (PHI_RNG_F   / (float)(W_ - 1)) + PHI_MIN_F;
    const float theta = gy * (THETA_RNG_F / (float)(H_ - 1)) + THETA_MIN_F;
    const float st = sinf(theta), ct = cosf(theta);
    const float d0 = st * cosf(phi);
    const float d1 = st * sinf(phi);
    const float d2 = ct;

    // B operand: lanes 0-15 -> (K0,K1) = (d0,d1); lanes 16-31 -> (K2,K3)=(d2,0)
    v2f bv;
    bv.x = hi ? d2 : d0;
    bv.y = hi ? 0.0f : d1;

    // A operands from pose rotation R (row-major 4x4)
    const float* P = pose + b * 16;
    const bool rowOK = (m < 3);
    const float R0 = rowOK ? P[m * 4 + 0] : 0.0f;
    const float R1 = rowOK ? P[m * 4 + 1] : 0.0f;
    const float R2 = rowOK ? P[m * 4 + 2] : 0.0f;

    // A_yin = R
    v2f aYin;
    aYin.x = hi ? R2 : R0;
    aYin.y = hi ? 0.0f : R1;
    // A_yang = R @ M_YANG ; M = [[-1,0,0],[0,0,1],[0,1,0]]
    //   cols: (-R[:,0], R[:,2], R[:,1])
    v2f aYang;
    aYang.x = hi ? R1 : -R0;
    aYang.y = hi ? 0.0f : R2;

    v8f cz = {};
    // (neg_a, A, neg_b, B, c_mod, C, reuse_a, reuse_b)
    v8f dYin  = __builtin_amdgcn_wmma_f32_16x16x4_f32(false, aYin,  false, bv,
                                                      (short)0, cz, false, false);
    v8f dYang = __builtin_amdgcn_wmma_f32_16x16x4_f32(false, aYang, false, bv,
                                                      (short)0, cz, false, false);

    if (!hi) {   // lanes 0-15 hold rows 0..2 for N = lane
        float* ry = rot + ((size_t)(b * 2 + 0) * 3) * HW_ + n;
        ry[0 * HW_] = dYin[0];
        ry[1 * HW_] = dYin[1];
        ry[2 * HW_] = dYin[2];
        float* rg = rot + ((size_t)(b * 2 + 1) * 3) * HW_ + n;
        rg[0 * HW_] = dYang[0];
        rg[1 * HW_] = dYang[1];
        rg[2 * HW_] = dYang[2];
    }
}

// ---------------------------------------------------------------------------
// Bilinear setup for one projected 3-D point in one hemisphere view.
// Produces 4 (element-offset, weight) pairs (validity folded into weight)
// and the "ones-channel" weight sum.
// ---------------------------------------------------------------------------
__device__ __forceinline__
void grid_setup(float x, float y, float z, int sBit,
                int* __restrict__ offs, float* __restrict__ wts,
                float& wsum)
{
    const float r  = sqrtf(x * x + y * y + z * z + 1e-12f);
    const float zi = fminf(fmaxf(z / r, -1.0f), 1.0f);
    const float theta = acosf(zi);
    const float phi   = atan2f(y, x);
    const float sc = r / fmaxf(r, 0.001f);          // CLAMP_MIN_DEPTH
    const float sx = (phi   - PHI_MIN_F)   * ((float)(W_ - 1) / PHI_RNG_F)   * sc;
    const float sy = (theta - THETA_MIN_F) * ((float)(H_ - 1) / THETA_RNG_F) * sc;

    const float x0f = floorf(sx), y0f = floorf(sy);
    const int x0 = (int)x0f, y0 = (int)y0f;
    const int x1 = x0 + 1,   y1 = y0 + 1;
    const float tx = sx - x0f, ty = sy - y0f;

    const float vx0 = (x0 >= 0 && x0 < W_) ? 1.0f : 0.0f;
    const float vx1 = (x1 >= 0 && x1 < W_) ? 1.0f : 0.0f;
    const float vy0 = (y0 >= 0 && y0 < H_) ? 1.0f : 0.0f;
    const float vy1 = (y1 >= 0 && y1 < H_) ? 1.0f : 0.0f;

    const float w00 = (1.0f - tx) * (1.0f - ty) * vx0 * vy0;
    const float w10 = tx          * (1.0f - ty) * vx1 * vy0;
    const float w01 = (1.0f - tx) * ty          * vx0 * vy1;
    const float w11 = tx          * ty          * vx1 * vy1;

    const int cx0 = min(max(x0, 0), W_ - 1);
    const int cx1 = min(max(x1, 0), W_ - 1);
    const int cy0 = min(max(y0, 0), H_ - 1);
    const int cy1 = min(max(y1, 0), H_ - 1);

    offs[0] = (cy0 * W_ + cx0) * 2 + sBit;  wts[0] = w00;
    offs[1] = (cy0 * W_ + cx1) * 2 + sBit;  wts[1] = w10;
    offs[2] = (cy1 * W_ + cx0) * 2 + sBit;  wts[2] = w01;
    offs[3] = (cy1 * W_ + cx1) * 2 + sBit;  wts[3] = w11;
    wsum = w00 + w10 + w01 + w11;
}

// ---------------------------------------------------------------------------
// Kernel 2: per-(b,d,h,w) geometry once, then a 128-channel gather loop.
// Gathers hit the L2-resident 17 MB feature map; the 512 MiB output is
// written with non-temporal float2 stores so it streams past L2.
// ---------------------------------------------------------------------------
__global__ __launch_bounds__(256)
void cross_warp_kernel(const float* __restrict__ feat,   // (B,C,H,W,2)
                       const float* __restrict__ pose,   // (B,4,4)
                       const float* __restrict__ depth,  // (B,D,H,W)
                       const float* __restrict__ rot,    // ws: (B,2,3,HW)
                       float* __restrict__ out)          // (B,C,D,H,W,2)
{
    const int tid = blockIdx.x * 256 + threadIdx.x;  // = ((b*D+d)*H+h)*W + w
    const int w = tid % W_;
    const int h = (tid / W_) % H_;
    const int d = (tid / HW_) % D_;
    const int b = tid / (D_ * HW_);
    const int n = h * W_ + w;

    const float dep = depth[tid];
    const float* P = pose + b * 16;
    const float t0 = P[3], t1 = P[7], t2 = P[11];

    const float* ry = rot + (size_t)(b * 2 + 0) * 3 * HW_ + n;
    const float* rg = rot + (size_t)(b * 2 + 1) * 3 * HW_ + n;
    const float pYx = fmaf(ry[0 * HW_], dep, t0);
    const float pYy = fmaf(ry[1 * HW_], dep, t1);
    const float pYz = fmaf(ry[2 * HW_], dep, t2);
    const float pGx = fmaf(rg[0 * HW_], dep, t0);
    const float pGy = fmaf(rg[1 * HW_], dep, t1);
    const float pGz = fmaf(rg[2 * HW_], dep, t2);

    int   offs[16];
    float wts[16];
    float wTifi, wTafi, wTifa, wTafa;
    // yin point seen in yin view (samples feat[...,0])
    grid_setup( pYx, pYy, pYz, 0, offs + 0,  wts + 0,  wTifi);
    // yin point seen in yang view: M@p = (-x, z, y)  (samples feat[...,1])
    grid_setup(-pYx, pYz, pYy, 1, offs + 4,  wts + 4,  wTafi);
    // yang point seen in yin view (samples feat[...,0])
    grid_setup( pGx, pGy, pGz, 0, offs + 8,  wts + 8,  wTifa);
    // yang point seen in yang view (samples feat[...,1])
    grid_setup(-pGx, pGz, pGy, 1, offs + 12, wts + 12, wTafa);

    const float wyin  = wTifi / (wTifi + wTafi + 1e-8f);
    const float wyang = wTafa / (wTifa + wTafa + 1e-8f);

    // Pre-combine blend weight * bilinear weight (validity already folded in)
#pragma unroll
    for (int j = 0; j < 4; ++j) {
        wts[j]      *= wyin;
        wts[4 + j]  *= (1.0f - wyin);
        wts[8 + j]  *= (1.0f - wyang);
        wts[12 + j] *= wyang;
    }

    const float* fb = feat + (size_t)(b * C_) * (HW_ * 2);
    float* op = out + ((((size_t)(b * C_) * D_ + d) * H_ + h) * W_ + w) * 2;
    const size_t fstride = (size_t)HW_ * 2;           // per channel
    const size_t ostride = (size_t)D_ * HW_ * 2;      // per channel

#pragma unroll 2
    for (int c = 0; c < C_; ++c) {
        float vy = 0.0f, vg = 0.0f;
#pragma unroll
        for (int j = 0; j < 8; ++j)  vy = fmaf(wts[j], fb[offs[j]], vy);
#pragma unroll
        for (int j = 8; j < 16; ++j) vg = fmaf(wts[j], fb[offs[j]], vg);
        f2 o; o.x = vy; o.y = vg;
        __builtin_nontemporal_store(o, (f2*)op);      // stream past L2
        fb += fstride;
        op += ostride;
    }
}

// ---------------------------------------------------------------------------
extern "C" void kernel_launch(void* const* d_in, const int* in_sizes, int n_in,
                              void* d_out, int out_size, void* d_ws, size_t ws_size,
                              hipStream_t stream)
{
    const float* feature1 = (const float*)d_in[0];  // (B,C,H,W,2)
    // d_in[1] = intrinsics (unused by the reference computation)
    const float* pose  = (const float*)d_in[2];     // (B,4,4)
    const float* depth = (const float*)d_in[3];     // (B,D,H,W)
    float* out = (float*)d_out;                     // (B,C,D,H,W,2)
    float* rot = (float*)d_ws;                      // (B,2,3,HW) = 393 KB

    // Kernel 1: WMMA rotation of the shared direction field (1024 waves).
    rot_dirs_wmma_kernel<<<B_ * (HW_ / 16), 32, 0, stream>>>(pose, rot);

    // Kernel 2: one thread per (b,d,h,w), 128-channel gather/blend loop.
    const int total = B_ * D_ * H_ * W_;            // 524288
    cross_warp_kernel<<<total / 256, 256, 0, stream>>>(feature1, pose, depth,
                                                       rot, out);
}